// TorchSolver_34703335752028
// MI455X (gfx1250) — compile-verified
//
#include <hip/hip_runtime.h>

// Problem dimensions (match reference setup_inputs()).
#define N_GENOMES 4000
#define N_SAMPLES 2048   // s: columns per row
#define N_GENES   28000  // m: one block per gene
#define N_SEQS    16000  // k: output rows

#define TPB 256                        // 8 wave32 waves per block
#define F4_PER_ROW (N_SAMPLES / 4)     // 512 float4 per row
#define F4_ITERS   (F4_PER_ROW / TPB)  // 2 iterations of 16B async copies per row

// Async copy 16 bytes from global memory to LDS (gfx1250 GLOBAL_LOAD_ASYNC_TO_LDS_B128).
// ldsByteAddr is the 32-bit LDS byte offset (low 32 bits of a generic LDS pointer).
__device__ __forceinline__ void async_g2l_b128(unsigned int ldsByteAddr, const void* gaddr) {
    asm volatile("global_load_async_to_lds_b128 %0, %1, off"
                 :: "v"(ldsByteAddr), "v"(gaddr)
                 : "memory");
}

__device__ __forceinline__ void wait_asynccnt0() {
    asm volatile("s_wait_asynccnt 0" ::: "memory");
}

__device__ __forceinline__ unsigned int lds_addr_of(const void* p) {
    // Generic pointers to LDS carry the LDS byte offset in their low 32 bits
    return (unsigned int)(unsigned long long)p;
}

// Hardware exp2: v_exp_f32. Argument range here is (0, 1], well-conditioned.
__device__ __forceinline__ float hw_exp2(float x) {
    return __builtin_amdgcn_exp2f(x);
}

__global__ __launch_bounds__(TPB) void gene_gather_scatter_kernel(
    const float* __restrict__ A,          // (N_GENOMES, N_SAMPLES)
    const float* __restrict__ B,          // (N_GENOMES, N_SAMPLES)
    const float* __restrict__ bias,       // (N_SEQS,)
    const float* __restrict__ pos,        // (N_GENES,)
    const int*   __restrict__ genome_idx, // (N_GENES,)
    const int*   __restrict__ seq_idx,    // (N_GENES,)
    float*       __restrict__ out)        // (N_SEQS, N_SAMPLES), pre-zeroed
{
    __shared__ __attribute__((aligned(16))) float sA[N_SAMPLES];
    __shared__ __attribute__((aligned(16))) float sB[N_SAMPLES];

    const int g  = blockIdx.x;            // gene id (uniform per block)
    const int gi = genome_idx[g];         // uniform -> scalar loads
    const int sq = seq_idx[g];
    const float p  = pos[g];
    const float bs = bias[sq];

    const float* __restrict__ aRow = A + (size_t)gi * N_SAMPLES;
    const float* __restrict__ bRow = B + (size_t)gi * N_SAMPLES;

    const int t = threadIdx.x;

    // Stage both rows into LDS with async 128-bit copies (ASYNCcnt-tracked).
#pragma unroll
    for (int it = 0; it < F4_ITERS; ++it) {
        const int e4 = t + it * TPB;            // float4 index within the row
        async_g2l_b128(lds_addr_of(&sA[e4 * 4]), (const void*)(aRow + e4 * 4));
        async_g2l_b128(lds_addr_of(&sB[e4 * 4]), (const void*)(bRow + e4 * 4));
    }
    wait_asynccnt0();     // my wave's async copies landed in LDS
    __syncthreads();      // everyone's copies landed

    float* __restrict__ outRow = out + (size_t)sq * N_SAMPLES;

#pragma unroll
    for (int it = 0; it < F4_ITERS; ++it) {
        const int base = (t + it * TPB) * 4;
        const float4 a = *(const float4*)&sA[base];   // ds_load_b128
        const float4 b = *(const float4*)&sB[base];

        // val = bias[sq] * a * 2^(1 - p*b)   (v_exp_f32)
        float4 v;
        v.x = bs * a.x * hw_exp2(__builtin_fmaf(-p, b.x, 1.0f));
        v.y = bs * a.y * hw_exp2(__builtin_fmaf(-p, b.y, 1.0f));
        v.z = bs * a.z * hw_exp2(__builtin_fmaf(-p, b.z, 1.0f));
        v.w = bs * a.w * hw_exp2(__builtin_fmaf(-p, b.w, 1.0f));

        // Segment-sum via no-return f32 atomics (low contention: ~1.75 genes/seq).
        atomicAdd(&outRow[base + 0], v.x);
        atomicAdd(&outRow[base + 1], v.y);
        atomicAdd(&outRow[base + 2], v.z);
        atomicAdd(&outRow[base + 3], v.w);
    }
}

extern "C" void kernel_launch(void* const* d_in, const int* in_sizes, int n_in,
                              void* d_out, int out_size, void* d_ws, size_t ws_size,
                              hipStream_t stream) {
    (void)in_sizes; (void)n_in; (void)d_ws; (void)ws_size;

    const float* A          = (const float*)d_in[0];
    const float* B          = (const float*)d_in[1];
    const float* bias       = (const float*)d_in[2];
    const float* pos        = (const float*)d_in[3];
    const int*   genome_idx = (const int*)d_in[4];
    const int*   seq_idx    = (const int*)d_in[5];
    float*       out        = (float*)d_out;

    // Output accumulates via atomics -> must start at zero every call
    // (graph-capturable memset node).
    (void)hipMemsetAsync(out, 0, (size_t)out_size * sizeof(float), stream);

    gene_gather_scatter_kernel<<<N_GENES, TPB, 0, stream>>>(
        A, B, bias, pos, genome_idx, seq_idx, out);
}